// WavePlaneField_63359357550929
// MI455X (gfx1250) — compile-verified
//
#include <hip/hip_runtime.h>
#include <hip/hip_bf16.h>

// ---------------------------------------------------------------------------
// WavePlaneField on MI455X (gfx1250).
//
// IDWT level (periodized 24-tap coif4) as a matrix sandwich:
//     out = A_{2h} @ [[ll, hl],[lh, hh]] @ A_{2w}^T
// with A_n a banded circulant (exactly one tap per entry since n >= 32 > 24).
// GEMMs run on V_WMMA_F32_16X16X4_F32 with 4-wave workgroups, 32x32 block
// tiles, and K-panels staged global->LDS via GLOBAL_LOAD_ASYNC_TO_LDS_B128
// (ASYNCcnt + s_wait_asynccnt), LDS rows padded to 36 dwords (16B-aligned
// B128 stores, bank-conflict-free strided reads).
// Final planes are channel-last [H,W,C]: the 6-plane bilinear gather is
// L2-resident (31.5 MB << 192 MB L2) and fully coalesced over 32 channels.
// ---------------------------------------------------------------------------

typedef __attribute__((ext_vector_type(2))) float v2f;
typedef __attribute__((ext_vector_type(8))) float v8f;

__constant__ float DEC_LO_C[24] = {
    -1.7849850030882614e-06f, -3.2596802368833675e-06f,  3.1229875865345646e-05f,
     6.233903446100713e-05f,  -0.00025997455248771324f, -0.0005890207562443383f,
     0.0012665619292989445f,   0.003751436157278457f,   -0.00565828668661072f,
    -0.015211731527946259f,    0.025082261844864097f,    0.03933442712333749f,
    -0.09622044203398798f,    -0.06662747426342504f,     0.4343860564914685f,
     0.782238930920499f,       0.41530840703043026f,    -0.05607731331675481f,
    -0.08126669968087875f,     0.026682300156053072f,    0.016068943964776348f,
    -0.0073461663276420935f,  -0.0016294920126017326f,   0.0008923136685823146f
};

// Build the n x n synthesis matrix A_n (REC_LO | REC_HI upsample-conv matrix).
__global__ void build_syn_matrix(float* __restrict__ A, int n) {
    int tid = blockIdx.x * blockDim.x + threadIdx.x;
    if (tid >= n * n) return;
    int i = tid / n, j = tid % n;
    int h = n >> 1;
    float v = 0.0f;
    if (j < h) {
        int k = (i - 2 * j + n) % n;                 // i-2j in (-n, n)
        if (k < 24) v = DEC_LO_C[23 - k];            // REC_LO[k] = DEC_LO[23-k]
    } else {
        int jj = j - h;
        int k = (i - 2 * jj + n) % n;
        if (k < 24) {                                // REC_HI[k] = DEC_LO[k]*(-1)^k
            float d = DEC_LO_C[k];
            v = (k & 1) ? -d : d;
        }
    }
    A[tid] = v;
}

// Assemble X [C, 2h, 2w] = [[ll, hl], [lh, hh]] from ll [C,h,w] and yh [C,3,h,w]
// (yh subband order: 0=lh, 1=hl, 2=hh).
__global__ void assemble_level(const float* __restrict__ ll,
                               const float* __restrict__ yh,
                               float* __restrict__ X, int h, int w) {
    int tid = blockIdx.x * blockDim.x + threadIdx.x;
    int H2 = 2 * h, W2 = 2 * w;
    int total = 32 * H2 * W2;
    if (tid >= total) return;
    int c = tid / (H2 * W2);
    int rem = tid % (H2 * W2);
    int i = rem / W2, j = rem % W2;
    int hw = h * w;
    float v;
    if (i < h) {
        if (j < w) v = ll[c * hw + i * w + j];                         // ll
        else       v = yh[(c * 3 + 1) * hw + i * w + (j - w)];         // hl
    } else {
        if (j < w) v = yh[(c * 3 + 0) * hw + (i - h) * w + j];         // lh
        else       v = yh[(c * 3 + 2) * hw + (i - h) * w + (j - w)];   // hh
    }
    X[tid] = v;
}

// ---------------------------------------------------------------------------
// Unified batched WMMA GEMM with async global->LDS staging.
//   TRANSB=false:  Z[c] = A (batched? strideA) @ B[c]      (B[k][j])
//   TRANSB=true :  Z[c] = A[c] @ B^T                       (B[j][k])
// Block = 128 threads (4 wave32), 32x32 output tile (2x2 waves of 16x16).
// K-panels (32 wide) of both operands are staged in LDS per K-block via
// GLOBAL_LOAD_ASYNC_TO_LDS_B128; panels padded to 36-dword row stride.
// ---------------------------------------------------------------------------
template <bool TRANSB, bool CLAST>
__global__ void gemm_wmma_lds(const float* __restrict__ Abase, long strideA, int lda,
                              const float* __restrict__ Bbase, long strideB, int ldb,
                              float* __restrict__ Zbase, long strideZ, int ldz,
                              int K) {
    __shared__ float As[32 * 36];
    __shared__ float Bs[32 * 36];

    const int c    = blockIdx.z;
    const int rb   = blockIdx.y * 32;
    const int jb   = blockIdx.x * 32;
    const int t    = threadIdx.x;        // 0..127
    const int lane = t & 31;
    const int w    = t >> 5;             // wave id 0..3
    const int wi   = w >> 1, wj = w & 1; // 2x2 wave tiling
    const int hf   = lane >> 4, l16 = lane & 15;

    const float* A = Abase + (long)c * strideA;
    const float* B = Bbase + (long)c * strideB;

    const int rloc = wi * 16 + l16;
    const int cloc = wj * 16 + l16;

    v8f acc = {};

    for (int kb = 0; kb < K; kb += 32) {
        __syncthreads();   // previous K-block's LDS readers done
        // ---- stage 32-row x 32-col panels of A and B (256 b128 chunks each,
        //      2 chunks per thread per panel), async global -> LDS ----
#pragma unroll
        for (int s = 0; s < 2; ++s) {
            int chunk = t + s * 128;      // 0..255
            int row   = chunk >> 3;       // 0..31
            int q4    = (chunk & 7) * 4;  // dword offset 0..28 (16B aligned)
            const float* ga = A + (long)(rb + row) * lda + kb + q4;
            unsigned la = (unsigned)(size_t)&As[row * 36 + q4];
            asm volatile("global_load_async_to_lds_b128 %0, %1, off"
                         :: "v"(la), "v"((unsigned long long)(size_t)ga)
                         : "memory");
            const float* gb = TRANSB ? B + (long)(jb + row) * ldb + kb + q4
                                     : B + (long)(kb + row) * ldb + jb + q4;
            unsigned lb = (unsigned)(size_t)&Bs[row * 36 + q4];
            asm volatile("global_load_async_to_lds_b128 %0, %1, off"
                         :: "v"(lb), "v"((unsigned long long)(size_t)gb)
                         : "memory");
        }
        asm volatile("s_wait_asynccnt 0" ::: "memory");
        __syncthreads();

        // ---- 8 x V_WMMA_F32_16X16X4_F32 on the staged panels ----
#pragma unroll
        for (int ks = 0; ks < 32; ks += 4) {
            int k0 = ks + 2 * hf;         // A lane layout: lane=(k/2)*16+m
            v2f a, b;
            a.x = As[rloc * 36 + k0];
            a.y = As[rloc * 36 + k0 + 1];
            if (TRANSB) {
                b.x = Bs[cloc * 36 + k0];
                b.y = Bs[cloc * 36 + k0 + 1];
            } else {
                b.x = Bs[k0 * 36 + cloc];
                b.y = Bs[(k0 + 1) * 36 + cloc];
            }
            acc = __builtin_amdgcn_wmma_f32_16x16x4_f32(
                false, a, false, b, (short)0, acc, false, false);
        }
    }

    const int col = jb + wj * 16 + l16;
#pragma unroll
    for (int v = 0; v < 8; ++v) {
        int r = rb + wi * 16 + v + 8 * hf;   // C/D layout: vgpr=m%8, lanehi=m/8
        if (CLAST)
            Zbase[((long)r * ldz + col) * 32 + c] = acc[v];   // [H, W, C]
        else
            Zbase[(long)c * strideZ + (long)r * ldz + col] = acc[v];
    }
}

// Bilinear sample + product over 6 planes.  Thread = (point, channel); the 32
// threads of a point read 128 contiguous bytes per corner (channel-last).
__global__ void sample_product(const float* __restrict__ pts,
                               const float* __restrict__ ts,
                               const float* __restrict__ pl0,
                               const float* __restrict__ pl1,
                               const float* __restrict__ pl2,
                               const float* __restrict__ pl3,
                               const float* __restrict__ pl4,
                               const float* __restrict__ pl5,
                               float* __restrict__ out, int N) {
    long tid = (long)blockIdx.x * blockDim.x + threadIdx.x;
    int pt = (int)(tid >> 5);
    int c  = (int)(tid & 31);
    if (pt >= N) return;

    const float inv_b = 1.0f / 1.3f;
    float crd[4];
    crd[0] = -pts[pt * 3 + 0] * inv_b;   // pts_n = -pts / BOUNDS
    crd[1] = -pts[pt * 3 + 1] * inv_b;
    crd[2] = -pts[pt * 3 + 2] * inv_b;
    crd[3] = ts[pt] * 2.0f - 1.0f;

    const float* planes[6] = {pl0, pl1, pl2, pl3, pl4, pl5};
    const int Ws[6] = {256, 256, 64, 256, 64, 64};     // H == 256 for all
    const int QI[6] = {0, 0, 3, 1, 3, 3};              // coords[:,0] -> W (x)
    const int RI[6] = {1, 2, 0, 2, 1, 2};              // coords[:,1] -> H (y)

    float prod = 1.0f;
#pragma unroll
    for (int p = 0; p < 6; ++p) {
        const int W = Ws[p], H = 256;
        float xg = crd[QI[p]], yg = crd[RI[p]];
        float x = fminf(fmaxf((xg + 1.0f) * 0.5f * (float)(W - 1), 0.0f), (float)(W - 1));
        float y = fminf(fmaxf((yg + 1.0f) * 0.5f * (float)(H - 1), 0.0f), (float)(H - 1));
        float x0f = fminf(floorf(x), (float)(W - 2));
        float y0f = fminf(floorf(y), (float)(H - 2));
        float wx = x - x0f, wy = y - y0f;
        int x0 = (int)x0f, y0 = (int)y0f;
        const float* pl = planes[p];
        long b00 = ((long)y0 * W + x0) * 32 + c;
        long rstr = (long)W * 32;
        __builtin_prefetch(pl + b00 + rstr, 0, 0);     // global_prefetch_b8
        float v00 = pl[b00];
        float v01 = pl[b00 + 32];
        float v10 = pl[b00 + rstr];
        float v11 = pl[b00 + rstr + 32];
        float val = v00 * (1.0f - wy) * (1.0f - wx)
                  + v01 * (1.0f - wy) * wx
                  + v10 * wy * (1.0f - wx)
                  + v11 * wy * wx;
        prod *= val;
    }
    out[(long)pt * 32 + c] = prod;
}

extern "C" void kernel_launch(void* const* d_in, const int* in_sizes, int n_in,
                              void* d_out, int out_size, void* d_ws, size_t ws_size,
                              hipStream_t stream) {
    const float* pts = (const float*)d_in[0];
    const float* ts  = (const float*)d_in[1];
    float* ws = (float*)d_ws;

    // Workspace layout (floats):
    float* A256 = ws;                       // 256*256 = 65536
    float* A128 = A256 + 65536;             // 128*128 = 16384
    float* A64  = A128 + 16384;             //  64* 64 =  4096
    float* A32  = A64  + 4096;              //  32* 32 =  1024
    float* SX   = A32  + 1024;              // 32*256*256 = 2097152
    float* SY   = SX   + 2097152;           // 2097152
    float* SZ   = SY   + 2097152;           // 32*128*128 = 524288
    float* PL   = SZ   + 524288;            // planes (channel-last)

    const int Wsarr[6] = {256, 256, 64, 256, 64, 64};
    float* plane_ptr[6];
    {
        long off = 0;
        for (int p = 0; p < 6; ++p) {
            plane_ptr[p] = PL + off;
            off += 32L * 256 * Wsarr[p];
        }
    }

    // 1. Build synthesis matrices (cheap, deterministic each call).
    build_syn_matrix<<<256, 256, 0, stream>>>(A256, 256);
    build_syn_matrix<<< 64, 256, 0, stream>>>(A128, 128);
    build_syn_matrix<<< 16, 256, 0, stream>>>(A64,  64);
    build_syn_matrix<<<  4, 256, 0, stream>>>(A32,  32);

    auto Amat = [&](int n) -> float* {
        return n == 256 ? A256 : n == 128 ? A128 : n == 64 ? A64 : A32;
    };

    // 2. Per plane: coarse IDWT level then fine IDWT level, all via WMMA GEMMs.
    for (int p = 0; p < 6; ++p) {
        const float* yl  = (const float*)d_in[2 + 3 * p];   // [C, h1, w1]
        const float* yha = (const float*)d_in[3 + 3 * p];   // [C, 3, h2, w2]
        const float* yhb = (const float*)d_in[4 + 3 * p];   // [C, 3, h1, w1]
        const int H = 256, W = Wsarr[p];
        const int h1 = H / 4, w1 = W / 4;
        const int n1 = 2 * h1, m1 = 2 * w1;     // coarse level output dims

        int tot1 = 32 * n1 * m1;
        long s1 = (long)n1 * m1;
        assemble_level<<<(tot1 + 255) / 256, 256, 0, stream>>>(yl, yhb, SX, h1, w1);
        // Y = A_{n1} @ X   (K = n1)
        gemm_wmma_lds<false, false><<<dim3(m1 / 32, n1 / 32, 32), 128, 0, stream>>>(
            Amat(n1), 0, n1, SX, s1, m1, SY, s1, m1, n1);
        // Z = Y @ A_{m1}^T (K = m1)
        gemm_wmma_lds<true, false><<<dim3(m1 / 32, n1 / 32, 32), 128, 0, stream>>>(
            SY, s1, m1, Amat(m1), 0, m1, SZ, s1, m1, m1);

        const int h2 = n1, w2 = m1;             // fine level input dims
        const int n2 = 2 * h2, m2 = 2 * w2;     // = H, W
        int tot2 = 32 * n2 * m2;
        long s2 = (long)n2 * m2;
        assemble_level<<<(tot2 + 255) / 256, 256, 0, stream>>>(SZ, yha, SX, h2, w2);
        gemm_wmma_lds<false, false><<<dim3(m2 / 32, n2 / 32, 32), 128, 0, stream>>>(
            Amat(n2), 0, n2, SX, s2, m2, SY, s2, m2, n2);
        // final stage stores channel-last [H, W, C]
        gemm_wmma_lds<true, true><<<dim3(m2 / 32, n2 / 32, 32), 128, 0, stream>>>(
            SY, s2, m2, Amat(m2), 0, m2, plane_ptr[p], 0, m2, m2);
    }

    // 3. Gather + product (L2-resident planes, coalesced channel-last reads).
    int N = in_sizes[0] / 3;
    long tot = (long)N * 32;
    int blocks = (int)((tot + 255) / 256);
    sample_product<<<blocks, 256, 0, stream>>>(pts, ts,
                                               plane_ptr[0], plane_ptr[1], plane_ptr[2],
                                               plane_ptr[3], plane_ptr[4], plane_ptr[5],
                                               (float*)d_out, N);
}